// Linear_59914793779559
// MI455X (gfx1250) — compile-verified
//
#include <hip/hip_runtime.h>
#include <stdint.h>

typedef __attribute__((ext_vector_type(2))) float v2f;
typedef __attribute__((ext_vector_type(4))) float v4f;
typedef __attribute__((ext_vector_type(8))) float v8f;

#define ROW_COLS 3328

// ---- CDNA5 async global->LDS staging (ASYNCcnt-tracked), with safe fallback ----
#if __has_builtin(__builtin_amdgcn_global_load_async_to_lds_b128) && \
    __has_builtin(__builtin_amdgcn_global_load_async_to_lds_b32)
#define HAVE_ASYNC_LDS 1
typedef int vi4 __attribute__((vector_size(16)));
typedef __attribute__((address_space(1))) vi4* gp_v4i;
typedef __attribute__((address_space(3))) vi4* lp_v4i;
typedef __attribute__((address_space(1))) int* gp_i32;
typedef __attribute__((address_space(3))) int* lp_i32;
#else
#define HAVE_ASYNC_LDS 0
#endif

__device__ __forceinline__ void copy_b128_to_lds(void* lds, const void* g) {
#if HAVE_ASYNC_LDS
  __builtin_amdgcn_global_load_async_to_lds_b128((gp_v4i)g, (lp_v4i)lds, 0, 0);
#else
  *(v4f*)lds = *(const v4f*)g;
#endif
}
__device__ __forceinline__ void copy_b32_to_lds(void* lds, const void* g) {
#if HAVE_ASYNC_LDS
  __builtin_amdgcn_global_load_async_to_lds_b32((gp_i32)g, (lp_i32)lds, 0, 0);
#else
  *(float*)lds = *(const float*)g;
#endif
}

#if HAVE_ASYNC_LDS
#if __has_builtin(__builtin_amdgcn_s_wait_asynccnt)
#define ASYNC_WAIT(n) __builtin_amdgcn_s_wait_asynccnt(n)
#else
#define ASYNC_WAIT(n) asm volatile("s_wait_asynccnt %0" ::"n"(n) : "memory")
#endif
#else
#define ASYNC_WAIT(n) ((void)0)
#endif

// Equivariant linear block:  Y[b, OFF + n*D + i] = pw * sum_u X[b, OFF + u*D + i] * W[u, n]
// Workgroup: 256 threads (8 waves). Tile: M=128 (16 rows/wave), N=64, all D sub-GEMMs.
// Two pipeline levels:
//  - LDS chunks double-buffered via async copies (ASYNCcnt).
//  - WMMA fragments double-buffered in VGPRs: load k-step ks+1 while multiplying ks,
//    so s_wait_dscnt covers loads issued a full WMMA group earlier.
template <int D, int MUL, int OFF, int KC>
__global__ __launch_bounds__(256)
void eqlin_kernel(const float* __restrict__ x, const float* __restrict__ w,
                  float* __restrict__ out, int B) {
  constexpr int M_TILE = 128;
  constexpr int N_TILE = 64;
  constexpr int XP = KC * D + 4;  // pitch = 4*odd dwords: 16B-aligned + conflict-free
  constexpr int WP = KC + 4;
  constexpr int ROW_F4 = (KC * D) / 4;
  constexpr int X_OPS = (M_TILE * ROW_F4) / 256;
  constexpr int W_OPS = (KC * N_TILE) / 256;
  constexpr int STAGE_OPS = X_OPS + W_OPS;
  constexpr int NCHUNK = MUL / KC;
  constexpr int KSTEP = KC / 4;

  __shared__ float Xs[2][M_TILE][XP];
  __shared__ float Wt[2][N_TILE][WP];  // Wt[b][n][k] = W[k0+k][n0+n]

  const int n0 = blockIdx.x * N_TILE;
  const int bm0 = blockIdx.y * M_TILE;
  const int tid = threadIdx.x;
  const int lane = tid & 31;
  const int wv = tid >> 5;
  const int ln = lane & 15;
  const int koff = (lane >> 4) << 1;  // 0 for lanes 0-15, 2 for lanes 16-31
  const int m0 = wv * 16;

  v8f acc[D][4];
#pragma unroll
  for (int i = 0; i < D; ++i)
#pragma unroll
    for (int n = 0; n < 4; ++n)
      acc[i][n] = v8f{};

  // Issue one stage of async copies (X tile b128, W tile transposed b32).
  auto stage = [&](int buf, int k0) {
    const float* xbase = x + OFF + k0 * D;
#pragma unroll
    for (int j = 0; j < X_OPS; ++j) {
      const int idx = tid + 256 * j;
      const int m = idx / ROW_F4;
      const int c = idx - m * ROW_F4;
      int gm = bm0 + m;
      if (gm >= B) gm = B - 1;  // clamp: tail rows load garbage, never stored
      copy_b128_to_lds(&Xs[buf][m][c * 4], xbase + gm * ROW_COLS + c * 4);
    }
#pragma unroll
    for (int j = 0; j < W_OPS; ++j) {
      const int idx = tid + 256 * j;
      const int k = idx >> 6;  // N_TILE == 64
      const int n = idx & 63;
      copy_b32_to_lds(&Wt[buf][n][k], w + (k0 + k) * MUL + n0 + n);
    }
  };

  // Load all WMMA fragments (4 B-subtiles + D A-rows) for one k-step from LDS.
  auto load_frags = [&](int buf, int ks, v2f* bf, v2f* af) {
    const int kb = ks * 4 + koff;
#pragma unroll
    for (int n = 0; n < 4; ++n)
      bf[n] = *(const v2f*)&Wt[buf][n * 16 + ln][kb];  // 8B aligned
#pragma unroll
    for (int i = 0; i < D; ++i) {
      v2f a;
      a.x = Xs[buf][m0 + ln][(kb + 0) * D + i];
      a.y = Xs[buf][m0 + ln][(kb + 1) * D + i];
      af[i] = a;
    }
  };

  stage(0, 0);

  for (int c = 0; c < NCHUNK; ++c) {
    const int buf = c & 1;
    if (c + 1 < NCHUNK) {
      stage(buf ^ 1, (c + 1) * KC);  // prefetch next chunk into other buffer
      ASYNC_WAIT(STAGE_OPS);         // current chunk's copies have landed
    } else {
      ASYNC_WAIT(0);
    }
    __syncthreads();

    v2f bf[2][4];
    v2f af[2][D];
    load_frags(buf, 0, bf[0], af[0]);
#pragma unroll
    for (int ks = 0; ks < KSTEP; ++ks) {
      const int cur = ks & 1;
      if (ks + 1 < KSTEP)
        load_frags(buf, ks + 1, bf[cur ^ 1], af[cur ^ 1]);  // register double-buffer
#pragma unroll
      for (int i = 0; i < D; ++i)
#pragma unroll
        for (int n = 0; n < 4; ++n)
          acc[i][n] = __builtin_amdgcn_wmma_f32_16x16x4_f32(
              false, af[cur][i], false, bf[cur][n], (short)0, acc[i][n], false,
              false);
    }
    __syncthreads();  // all waves done reading `buf` before it is restaged
  }

  // ---- Epilogue: scale by path weight 1/sqrt(MUL), scatter to interleaved layout ----
  const float pw = (MUL == 512) ? 0.04419417382415922f : 0.0625f;
  const int mhi = (lane >> 4) << 3;  // +8 rows for lanes 16-31
#pragma unroll
  for (int r = 0; r < 8; ++r) {
    const int gm = bm0 + m0 + r + mhi;
    if (gm < B) {
      float* orow = out + gm * ROW_COLS + OFF;
#pragma unroll
      for (int n = 0; n < 4; ++n) {
        const int gn = n0 + n * 16 + ln;
#pragma unroll
        for (int i = 0; i < D; ++i)
          orow[gn * D + i] = pw * acc[i][n][r];
      }
    }
  }
}

extern "C" void kernel_launch(void* const* d_in, const int* in_sizes, int n_in,
                              void* d_out, int out_size, void* d_ws, size_t ws_size,
                              hipStream_t stream) {
  const float* x = (const float*)d_in[0];
  const float* w0 = (const float*)d_in[1];
  const float* w1 = (const float*)d_in[2];
  const float* w2 = (const float*)d_in[3];
  float* out = (float*)d_out;

  const int B = in_sizes[0] / ROW_COLS;
  const int mtiles = (B + 127) / 128;

  // block 0: 512x0e  (d=1, mul=512, offset 0),    N tiles = 512/64 = 8
  eqlin_kernel<1, 512, 0, 32><<<dim3(8, mtiles), 256, 0, stream>>>(x, w0, out, B);
  // block 1: 512x1o  (d=3, mul=512, offset 512),  N tiles = 8
  eqlin_kernel<3, 512, 512, 16><<<dim3(8, mtiles), 256, 0, stream>>>(x, w1, out, B);
  // block 2: 256x2e  (d=5, mul=256, offset 2048), N tiles = 4
  eqlin_kernel<5, 256, 2048, 8><<<dim3(4, mtiles), 256, 0, stream>>>(x, w2, out, B);
}